// GroupCorrFeatureScorer_8899172237742
// MI455X (gfx1250) — compile-verified
//
#include <hip/hip_runtime.h>

#define N_  9
#define C_  32
#define H_  192
#define W_  256
#define S_  9
#define G_  8
#define HW_  (H_*W_)          // 49152 pixels
#define HWS_ (H_*W_*S_)       // 442368 per-channel elements
#define NSTRIDE_ ((size_t)C_*HWS_)

typedef __attribute__((ext_vector_type(2))) float v2f;
typedef __attribute__((ext_vector_type(8))) float v8f;

// ---------------------------------------------------------------------------
// Prep: M = Wp^T Wp (32x32), v = Wp^T bp (32), k = bp.bp  -> ws[0..1056]
// ---------------------------------------------------------------------------
__global__ __launch_bounds__(256)
void gcfs_prep_kernel(const float* __restrict__ Wp, const float* __restrict__ bp,
                      float* __restrict__ ws) {
  int t = threadIdx.x;
  for (int idx = t; idx < 1024; idx += 256) {
    int d = idx >> 5, c = idx & 31;
    float s = 0.f;
    for (int e = 0; e < 32; ++e) s += Wp[e*32 + d] * Wp[e*32 + c];
    ws[idx] = s;
  }
  if (t < 32) {
    float s = 0.f;
    for (int e = 0; e < 32; ++e) s += bp[e] * Wp[e*32 + t];
    ws[1024 + t] = s;
  }
  if (t == 0) {
    float s = 0.f;
    for (int e = 0; e < 32; ++e) s += bp[e] * bp[e];
    ws[1056] = s;
  }
}

// ---------------------------------------------------------------------------
// Main: one wave32 = 16 pixels.
//   t[d,p]   = (Wp^T Wp) @ ref4   via v_wmma_f32_16x16x4_f32 (16 WMMAs/wave)
//   logits   = (t+v) . ref_s + v.ref4 + bp.bp  (D-layout elementwise + xor16)
//   softmax over s=9, then grouped correlation vs 8 source frames.
// ---------------------------------------------------------------------------
__global__ __launch_bounds__(256)
void gcfs_main_kernel(const float* __restrict__ F, const float* __restrict__ ws,
                      float* __restrict__ out) {
  const int tid  = blockIdx.x * blockDim.x + threadIdx.x;
  const int wave = tid >> 5;
  const int lane = threadIdx.x & 31;
  const int hi   = lane >> 4;          // lane-half (selects d rows / B K rows)
  const int l    = lane & 15;          // column index within tile (= pixel)
  const int hw   = wave * 16 + l;      // this lane's pixel (h*W + w)

  const float* Mm = ws;                // 32x32 row-major
  const float* v  = ws + 1024;
  const float  kk = ws[1056];
  const float* ref = F;                // frame n = 0

  // Pre-touch the 8 source frames for this pixel range (global_prefetch_b8).
  #pragma unroll
  for (int n = 0; n < 8; ++n)
    __builtin_prefetch(F + (size_t)(n + 1) * NSTRIDE_ + (size_t)hw * S_, 0, 0);

  // ---- t = M @ ref4 : two 16-row tiles, K=32 in 8 steps of 4 --------------
  v8f acc0 = {}; v8f acc1 = {};
  #pragma unroll
  for (int k = 0; k < 8; ++k) {
    const int cB = 4*k + 2*hi;         // K rows: lanes0-15 -> K0,K1 ; 16-31 -> K2,K3
    v2f b;
    b.x = ref[(size_t)(cB    ) * HWS_ + (size_t)hw * S_ + (S_/2)];
    b.y = ref[(size_t)(cB + 1) * HWS_ + (size_t)hw * S_ + (S_/2)];
    v2f a0, a1;                         // A: lane = row d, same K split
    a0.x = Mm[(     l) * 32 + cB];  a0.y = Mm[(     l) * 32 + cB + 1];
    a1.x = Mm[(16 + l) * 32 + cB];  a1.y = Mm[(16 + l) * 32 + cB + 1];
    acc0 = __builtin_amdgcn_wmma_f32_16x16x4_f32(false, a0, false, b,
                                                 (short)0, acc0, false, false);
    acc1 = __builtin_amdgcn_wmma_f32_16x16x4_f32(false, a1, false, b,
                                                 (short)0, acc1, false, false);
  }

  // tv = t + v  (D layout: VGPR r holds d = r + 8*hi, tile1 adds +16)
  float tv0[8], tv1[8];
  #pragma unroll
  for (int r = 0; r < 8; ++r) {
    tv0[r] = acc0[r] + v[     r + 8*hi];
    tv1[r] = acc1[r] + v[16 + r + 8*hi];
  }

  // cc = v . ref4 + bp.bp  (all 32 lanes compute for pixel l; halves agree)
  float cc = kk;
  for (int c = 0; c < 32; ++c)
    cc += v[c] * ref[(size_t)c * HWS_ + (size_t)hw * S_ + (S_/2)];

  // ---- logits + softmax ---------------------------------------------------
  const float rsC = 0.17677669529663687f;   // 1/sqrt(32)
  float att[S_];
  #pragma unroll
  for (int s = 0; s < S_; ++s) {
    float partial = 0.f;
    #pragma unroll
    for (int r = 0; r < 8; ++r) {
      float r0 = ref[(size_t)(     r + 8*hi) * HWS_ + (size_t)hw * S_ + s];
      float r1 = ref[(size_t)(16 + r + 8*hi) * HWS_ + (size_t)hw * S_ + s];
      partial += tv0[r] * r0 + tv1[r] * r1;
    }
    float full = partial + __shfl_xor(partial, 16, 32);  // join lane-halves
    att[s] = (full + cc) * rsC;
  }
  float m = att[0];
  #pragma unroll
  for (int s = 1; s < S_; ++s) m = fmaxf(m, att[s]);
  float sum = 0.f;
  #pragma unroll
  for (int s = 0; s < S_; ++s) { att[s] = __expf(att[s] - m); sum += att[s]; }
  const float inv = 1.f / sum;
  #pragma unroll
  for (int s = 0; s < S_; ++s) att[s] *= inv;

  // ---- grouped correlation: out[n,g] = sum_s att[s] sum_{c in g} ref*src --
  // Lane owns channels c = 16*t2 + 8*hi + r  -> groups {0,1,4,5} (hi=0)
  // or {2,3,6,7} (hi=1); group sums never cross lanes.
  float accO[8][4];
  #pragma unroll
  for (int n = 0; n < 8; ++n)
    #pragma unroll
    for (int a = 0; a < 4; ++a) accO[n][a] = 0.f;

  for (int s = 0; s < S_; ++s) {
    float wref[16];
    #pragma unroll
    for (int t2 = 0; t2 < 2; ++t2)
      #pragma unroll
      for (int r = 0; r < 8; ++r) {
        int c = 16*t2 + 8*hi + r;
        wref[t2*8 + r] = ref[(size_t)c * HWS_ + (size_t)hw * S_ + s] * att[s];
      }
    for (int n = 0; n < 8; ++n) {
      const float* src = F + (size_t)(n + 1) * NSTRIDE_;
      #pragma unroll
      for (int t2 = 0; t2 < 2; ++t2)
        #pragma unroll
        for (int r = 0; r < 8; ++r) {
          int c = 16*t2 + 8*hi + r;
          float sv = src[(size_t)c * HWS_ + (size_t)hw * S_ + s];
          accO[n][t2*2 + (r >> 2)] += wref[t2*8 + r] * sv;
        }
    }
  }

  // ---- store: a = t2*2 + (r>>2)  ->  g = 4*(a>>1) + 2*hi + (a&1) ----------
  #pragma unroll
  for (int n = 0; n < 8; ++n)
    #pragma unroll
    for (int a = 0; a < 4; ++a) {
      int g = 4*(a >> 1) + 2*hi + (a & 1);
      out[(size_t)(n * G_ + g) * HW_ + hw] = accO[n][a];
    }
}

// ---------------------------------------------------------------------------
extern "C" void kernel_launch(void* const* d_in, const int* in_sizes, int n_in,
                              void* d_out, int out_size, void* d_ws, size_t ws_size,
                              hipStream_t stream) {
  (void)in_sizes; (void)n_in; (void)out_size; (void)ws_size;
  const float* F  = (const float*)d_in[0];
  const float* Wp = (const float*)d_in[1];
  const float* bp = (const float*)d_in[2];
  float* ws  = (float*)d_ws;
  float* out = (float*)d_out;

  gcfs_prep_kernel<<<1, 256, 0, stream>>>(Wp, bp, ws);

  const int waves  = HW_ / 16;          // 3072 waves, 16 pixels each
  const int blocks = waves / 8;         // 256 threads = 8 wave32 per block
  gcfs_main_kernel<<<blocks, 256, 0, stream>>>(F, ws, out);
}